// ATG_GRU_16303695856129
// MI455X (gfx1250) — compile-verified
//
#include <hip/hip_runtime.h>
#include <math.h>

// ---------------- problem constants ----------------
constexpr int kB  = 32;   // batch
constexpr int kT  = 8;    // timesteps
constexpr int kN  = 150;  // nodes
constexpr int kF  = 16;   // input features
constexpr int kH  = 32;   // hidden
constexpr int kR  = 4;    // relations (FFT length)
constexpr int kE  = 16;   // embedding
constexpr int kNP = 160;  // nodes padded to 16
constexpr int kWX = kB * kF;   // 512
constexpr int kWH = kB * kH;   // 1024
static_assert(kH % 16 == 0 && kH / 16 == 2, "H tiling hardcoded");

typedef float v2f __attribute__((ext_vector_type(2)));
typedef float v8f __attribute__((ext_vector_type(8)));

static __device__ __forceinline__ v8f wmma4(v2f a, v2f b, v8f c) {
  // D = A(16x4 f32) * B(4x16 f32) + C(16x16 f32)
  return __builtin_amdgcn_wmma_f32_16x16x4_f32(false, a, false, b, (short)0, c,
                                               false, false);
}

// K-loop for one GEMM term with compile-time strides.
// Ap: row pointer (element Ap[k + 2g]).  Bp: column pointer (element
// Bp[(k+2g)*LDB]).  Constant LDB => all load offsets are immediates.
template <int LDB, int K>
static __device__ __forceinline__ v8f gemm_term(v8f acc,
                                                const float* __restrict__ Ap,
                                                const float* __restrict__ Bp,
                                                int g) {
  const float* __restrict__ Ag = Ap + 2 * g;
  const float* __restrict__ Bg = Bp + (size_t)(2 * g) * LDB;
#pragma unroll 8
  for (int k = 0; k < K; k += 4) {
    v2f a, b;
    a.x = Ag[k];
    a.y = Ag[k + 1];
    b.x = Bg[(size_t)k * LDB];
    b.y = Bg[(size_t)k * LDB + LDB];
    acc = wmma4(a, b, acc);
  }
  return acc;
}

// ============ elementwise / setup kernels ============

__global__ void k_zero(float* p, int n) {
  int i = blockIdx.x * blockDim.x + threadIdx.x;
  if (i < n) p[i] = 0.f;
}

// A[n,m,r] = relu( sum_e circconv(U[n,e,:], U[m,e,:])[r] )
__global__ void k_adj_tprod(const float* __restrict__ U, float* __restrict__ At) {
  int idx = blockIdx.x * blockDim.x + threadIdx.x;
  if (idx >= kN * kN) return;
  int n = idx / kN, m = idx % kN;
  float c[4] = {0.f, 0.f, 0.f, 0.f};
  for (int e = 0; e < kE; ++e) {
    float un[4], um[4];
#pragma unroll
    for (int r = 0; r < 4; ++r) {
      un[r] = U[((size_t)n * kE + e) * kR + r];
      um[r] = U[((size_t)m * kE + e) * kR + r];
    }
#pragma unroll
    for (int s = 0; s < 4; ++s)
#pragma unroll
      for (int r = 0; r < 4; ++r)
        c[(s + r) & 3] += un[s] * um[r];
  }
#pragma unroll
  for (int r = 0; r < 4; ++r)
    At[(size_t)idx * 4 + r] = c[r] > 0.f ? c[r] : 0.f;
}

// softmax over m (axis=1) for each (n, r), in place
__global__ void k_adj_softmax(float* At) {
  int idx = blockIdx.x * blockDim.x + threadIdx.x;
  if (idx >= kN * kR) return;
  int n = idx / kR, r = idx % kR;
  float mx = -1e30f;
  for (int m = 0; m < kN; ++m) mx = fmaxf(mx, At[((size_t)n * kN + m) * 4 + r]);
  float s = 0.f;
  for (int m = 0; m < kN; ++m) s += __expf(At[((size_t)n * kN + m) * 4 + r] - mx);
  float inv = 1.f / s;
  for (int m = 0; m < kN; ++m) {
    size_t o = ((size_t)n * kN + m) * 4 + r;
    At[o] = __expf(At[o] - mx) * inv;
  }
}

// faces of adj, zero-padded to kNP x kNP. plane order: f0, f2, f1_re, f1_im
__global__ void k_adj_fft(const float* __restrict__ At, float* __restrict__ Af) {
  int idx = blockIdx.x * blockDim.x + threadIdx.x;
  if (idx >= kNP * kNP) return;
  int n = idx / kNP, m = idx % kNP;
  float f0 = 0.f, f2 = 0.f, fr = 0.f, fi = 0.f;
  if (n < kN && m < kN) {
    const float* a = At + ((size_t)n * kN + m) * 4;
    f0 = a[0] + a[1] + a[2] + a[3];
    f2 = a[0] - a[1] + a[2] - a[3];
    fr = a[0] - a[2];
    fi = a[3] - a[1];
  }
  size_t P = (size_t)kNP * kNP;
  Af[idx] = f0; Af[P + idx] = f2; Af[2 * P + idx] = fr; Af[3 * P + idx] = fi;
}

// faces of a weight tensor [K,H,R] -> 4 planes of K*H
__global__ void k_w_fft(const float* __restrict__ W, float* __restrict__ Wf, int KH) {
  int idx = blockIdx.x * blockDim.x + threadIdx.x;
  if (idx >= KH) return;
  const float* w = W + (size_t)idx * 4;
  float x0 = w[0], x1 = w[1], x2 = w[2], x3 = w[3];
  size_t P = (size_t)KH;
  Wf[idx] = x0 + x1 + x2 + x3;
  Wf[P + idx] = x0 - x1 + x2 - x3;
  Wf[2 * P + idx] = x0 - x2;
  Wf[3 * P + idx] = x3 - x1;
}

// faces of inputs[:, t] -> planes [kNP][kWX], rows >= kN zeroed
__global__ void k_in_fft(const float* __restrict__ inp, int t, float* __restrict__ Xf) {
  int idx = blockIdx.x * blockDim.x + threadIdx.x;
  if (idx >= kNP * kWX) return;
  int n = idx / kWX, c = idx % kWX;
  int b = c / kF, f = c % kF;
  float x0 = 0.f, x1 = 0.f, x2 = 0.f, x3 = 0.f;
  if (n < kN) {
    const float* p = inp + ((((size_t)b * kT + t) * kN + n) * kF + f) * kR;
    x0 = p[0]; x1 = p[1]; x2 = p[2]; x3 = p[3];
  }
  size_t P = (size_t)kNP * kWX;
  Xf[idx] = x0 + x1 + x2 + x3;
  Xf[P + idx] = x0 - x1 + x2 - x3;
  Xf[2 * P + idx] = x0 - x2;
  Xf[3 * P + idx] = x3 - x1;
}

// faces of a time-domain buffer [kNP][Wc][4]
__global__ void k_s_fft(const float* __restrict__ S, float* __restrict__ Sf, int Wc) {
  int idx = blockIdx.x * blockDim.x + threadIdx.x;
  if (idx >= kNP * Wc) return;
  const float* p = S + (size_t)idx * 4;
  float x0 = p[0], x1 = p[1], x2 = p[2], x3 = p[3];
  size_t P = (size_t)kNP * Wc;
  Sf[idx] = x0 + x1 + x2 + x3;
  Sf[P + idx] = x0 - x1 + x2 - x3;
  Sf[2 * P + idx] = x0 - x2;
  Sf[3 * P + idx] = x3 - x1;
}

// ============ WMMA GEMM kernels ============

// C_faces = Af @ X_faces, facewise (faces 0,2 real; face1 complex).
// grid: (kNP/16, W/128, 4 jobs), block: 256 (8 waves, wave = column tile)
template <int W>
__global__ __launch_bounds__(256) void k_agg_gemm(const float* __restrict__ A4,
                                                  const float* __restrict__ X4,
                                                  float* __restrict__ C4) {
  const int lane = threadIdx.x & 31;
  const int wave = threadIdx.x >> 5;
  const int mrow = lane & 15;
  const int g = lane >> 4;
  const int row0 = blockIdx.x << 4;
  const int col0 = ((blockIdx.y << 3) + wave) << 4;
  const int job = blockIdx.z;
  constexpr size_t AP = (size_t)kNP * kNP;
  constexpr size_t XP = (size_t)kNP * W;

  const float* Ar = A4 + (size_t)(row0 + mrow) * kNP;  // + plane*AP
  const float* Xc = X4 + col0 + mrow;                  // + plane*XP
  v8f p = {0.f, 0.f, 0.f, 0.f, 0.f, 0.f, 0.f, 0.f};
  v8f nn = {0.f, 0.f, 0.f, 0.f, 0.f, 0.f, 0.f, 0.f};
  if (job == 0) {
    p = gemm_term<W, kNP>(p, Ar, Xc, g);
  } else if (job == 1) {
    p = gemm_term<W, kNP>(p, Ar + AP, Xc + XP, g);
  } else if (job == 2) {  // Cr = Ar*Xr - Ai*Xi
    p  = gemm_term<W, kNP>(p,  Ar + 2 * AP, Xc + 2 * XP, g);
    nn = gemm_term<W, kNP>(nn, Ar + 3 * AP, Xc + 3 * XP, g);
  } else {                // Ci = Ar*Xi + Ai*Xr
    p = gemm_term<W, kNP>(p, Ar + 2 * AP, Xc + 3 * XP, g);
    p = gemm_term<W, kNP>(p, Ar + 3 * AP, Xc + 2 * XP, g);
  }
  float* Cp = C4 + (size_t)job * XP;
#pragma unroll
  for (int v = 0; v < 8; ++v)
    Cp[(size_t)(row0 + v + 8 * g) * W + col0 + mrow] = p[v] - nn[v];
}

// O_faces[n][b*H+o] = P_faces[n][b*Kp+f] @ Wa_faces[f][o]
//                   + Q_faces[n][b*Kq+f] @ Wb_faces[f][o]   (facewise complex)
// grid: (kNP/16, kB*(kH/16)/8, 4 jobs), block: 256 (8 waves)
template <int Wp, int Kp, int Wq, int Kq>
__global__ __launch_bounds__(256) void k_trans_gemm(
    const float* __restrict__ P4, const float* __restrict__ Wa4,
    const float* __restrict__ Q4, const float* __restrict__ Wb4,
    float* __restrict__ O4) {
  const int lane = threadIdx.x & 31;
  const int wave = threadIdx.x >> 5;
  const int mrow = lane & 15;
  const int g = lane >> 4;
  const int row0 = blockIdx.x << 4;
  const int ty = (blockIdx.y << 3) + wave;  // 0..63
  const int b = ty >> 1;
  const int col0 = (ty & 1) << 4;
  const int job = blockIdx.z;
  constexpr size_t Pps = (size_t)kNP * Wp, Qps = (size_t)kNP * Wq;
  constexpr size_t WaPs = (size_t)Kp * kH, WbPs = (size_t)Kq * kH;

  const float* Pr = P4 + (size_t)(row0 + mrow) * Wp + b * Kp;  // + plane*Pps
  const float* Qr = Q4 + (size_t)(row0 + mrow) * Wq + b * Kq;  // + plane*Qps
  const float* Wa = Wa4 + col0 + mrow;                         // + plane*WaPs
  const float* Wb = Wb4 + col0 + mrow;                         // + plane*WbPs

  v8f p = {0.f, 0.f, 0.f, 0.f, 0.f, 0.f, 0.f, 0.f};
  v8f nn = {0.f, 0.f, 0.f, 0.f, 0.f, 0.f, 0.f, 0.f};
  if (job == 0) {
    p = gemm_term<kH, Kp>(p, Pr, Wa, g);
    p = gemm_term<kH, Kq>(p, Qr, Wb, g);
  } else if (job == 1) {
    p = gemm_term<kH, Kp>(p, Pr + Pps, Wa + WaPs, g);
    p = gemm_term<kH, Kq>(p, Qr + Qps, Wb + WbPs, g);
  } else if (job == 2) {  // real part of face1
    p  = gemm_term<kH, Kp>(p,  Pr + 2 * Pps, Wa + 2 * WaPs, g);
    p  = gemm_term<kH, Kq>(p,  Qr + 2 * Qps, Wb + 2 * WbPs, g);
    nn = gemm_term<kH, Kp>(nn, Pr + 3 * Pps, Wa + 3 * WaPs, g);
    nn = gemm_term<kH, Kq>(nn, Qr + 3 * Qps, Wb + 3 * WbPs, g);
  } else {                // imaginary part of face1
    p = gemm_term<kH, Kp>(p, Pr + 2 * Pps, Wa + 3 * WaPs, g);
    p = gemm_term<kH, Kp>(p, Pr + 3 * Pps, Wa + 2 * WaPs, g);
    p = gemm_term<kH, Kq>(p, Qr + 2 * Qps, Wb + 3 * WbPs, g);
    p = gemm_term<kH, Kq>(p, Qr + 3 * Qps, Wb + 2 * WbPs, g);
  }
  float* Op = O4 + (size_t)job * kNP * kWH;
#pragma unroll
  for (int v = 0; v < 8; ++v)
    Op[(size_t)(row0 + v + 8 * g) * kWH + (size_t)b * kH + col0 + mrow] =
        p[v] - nn[v];
}

// ============ gate kernels (ifft + nonlinearity) ============

static __device__ __forceinline__ void ifft4(float f0, float f2, float fr,
                                             float fi, float* x) {
  x[0] = (f0 + f2 + 2.f * fr) * 0.25f;
  x[1] = (f0 - f2 - 2.f * fi) * 0.25f;
  x[2] = (f0 + f2 - 2.f * fr) * 0.25f;
  x[3] = (f0 - f2 + 2.f * fi) * 0.25f;
}

__global__ void k_gate_zr(const float* __restrict__ Zp, const float* __restrict__ Rp,
                          const float* __restrict__ Bias, const float* __restrict__ Hs,
                          float* __restrict__ Zt, float* __restrict__ Gt) {
  int idx = blockIdx.x * blockDim.x + threadIdx.x;
  if (idx >= kNP * kWH) return;
  int n = idx / kWH, c = idx % kWH;
  int h = c % kH;
  if (n >= kN) {
#pragma unroll
    for (int r = 0; r < 4; ++r) { Zt[(size_t)idx * 4 + r] = 0.f; Gt[(size_t)idx * 4 + r] = 0.f; }
    return;
  }
  size_t P = (size_t)kNP * kWH;
  float tz[4], tr[4];
  ifft4(Zp[idx], Zp[P + idx], Zp[2 * P + idx], Zp[3 * P + idx], tz);
  ifft4(Rp[idx], Rp[P + idx], Rp[2 * P + idx], Rp[3 * P + idx], tr);
#pragma unroll
  for (int r = 0; r < 4; ++r) {
    float z = 1.f / (1.f + __expf(-(tz[r] + Bias[(0 * kH + h) * kR + r])));
    float rg = 1.f / (1.f + __expf(-(tr[r] + Bias[(1 * kH + h) * kR + r])));
    float hs = Hs[(size_t)idx * 4 + r];
    Zt[(size_t)idx * 4 + r] = z;
    Gt[(size_t)idx * 4 + r] = rg * hs;
  }
}

__global__ void k_gate_h(const float* __restrict__ Hp, const float* __restrict__ Bias,
                         const float* __restrict__ Zt, const float* __restrict__ Hs,
                         float* __restrict__ Hnew, float* __restrict__ outp, int t) {
  int idx = blockIdx.x * blockDim.x + threadIdx.x;
  if (idx >= kNP * kWH) return;
  int n = idx / kWH, c = idx % kWH;
  int b = c / kH, h = c % kH;
  if (n >= kN) {
#pragma unroll
    for (int r = 0; r < 4; ++r) Hnew[(size_t)idx * 4 + r] = 0.f;
    return;
  }
  size_t P = (size_t)kNP * kWH;
  float th[4];
  ifft4(Hp[idx], Hp[P + idx], Hp[2 * P + idx], Hp[3 * P + idx], th);
#pragma unroll
  for (int r = 0; r < 4; ++r) {
    float ht = tanhf(th[r] + Bias[(2 * kH + h) * kR + r]);
    float z = Zt[(size_t)idx * 4 + r];
    float hn = z * Hs[(size_t)idx * 4 + r] + (1.f - z) * ht;
    Hnew[(size_t)idx * 4 + r] = hn;
    if (outp)
      outp[((((size_t)b * kT + t) * kN + n) * kH + h) * kR + r] = hn;
  }
}

// h_last slice: dst[b,n,h,r] = S[n][b*H+h][r]
__global__ void k_copy_hlast(const float* __restrict__ S, float* __restrict__ dst) {
  int idx = blockIdx.x * blockDim.x + threadIdx.x;
  if (idx >= kB * kN * kH) return;
  int h = idx % kH;
  int tmp = idx / kH;
  int n = tmp % kN, b = tmp / kN;
  const float* s = S + ((size_t)n * kWH + (size_t)b * kH + h) * 4;
  float* d = dst + (size_t)idx * 4;
#pragma unroll
  for (int r = 0; r < 4; ++r) d[r] = s[r];
}

// ============ host orchestration ============

struct Ctx {
  hipStream_t stream;
  const float* AF;
  float *HF, *AX, *AH, *GF, *AG, *ZP, *RP, *HP, *ZT, *GT;
};

static inline dim3 ewg(size_t n) { return dim3((unsigned)((n + 255) / 256)); }

// one TensorGRU cell (Win: face width of X, Kin: per-batch feature dim)
template <int Win, int Kin>
static void run_cell(const Ctx& c, const float* Xfaces, const float* Wxz,
                     const float* Wxr, const float* Wxh, const float* Whz,
                     const float* Whr, const float* bias, const float* Hs,
                     float* Hnew, float* outp, int t) {
  const size_t EW = (size_t)kNP * kWH;
  k_s_fft<<<ewg(EW), 256, 0, c.stream>>>(Hs, c.HF, kWH);
  k_agg_gemm<Win><<<dim3(kNP / 16, Win / 128, 4), 256, 0, c.stream>>>(
      c.AF, Xfaces, c.AX);
  k_agg_gemm<kWH><<<dim3(kNP / 16, kWH / 128, 4), 256, 0, c.stream>>>(
      c.AF, c.HF, c.AH);
  dim3 tg(kNP / 16, kB * (kH / 16) / 8, 4);
  k_trans_gemm<Win, Kin, kWH, kH><<<tg, 256, 0, c.stream>>>(c.AX, Wxz, c.AH,
                                                            Whz, c.ZP);
  k_trans_gemm<Win, Kin, kWH, kH><<<tg, 256, 0, c.stream>>>(c.AX, Wxr, c.AH,
                                                            Whr, c.RP);
  k_gate_zr<<<ewg(EW), 256, 0, c.stream>>>(c.ZP, c.RP, bias, Hs, c.ZT, c.GT);
  k_s_fft<<<ewg(EW), 256, 0, c.stream>>>(c.GT, c.GF, kWH);
  k_agg_gemm<kWH><<<dim3(kNP / 16, kWH / 128, 4), 256, 0, c.stream>>>(
      c.AF, c.GF, c.AG);
  // NOTE: faithful to source, H_tilda reuses Whr
  k_trans_gemm<Win, Kin, kWH, kH><<<tg, 256, 0, c.stream>>>(c.AX, Wxh, c.AG,
                                                            Whr, c.HP);
  k_gate_h<<<ewg(EW), 256, 0, c.stream>>>(c.HP, bias, c.ZT, Hs, Hnew, outp, t);
}

extern "C" void kernel_launch(void* const* d_in, const int* in_sizes, int n_in,
                              void* d_out, int out_size, void* d_ws, size_t ws_size,
                              hipStream_t stream) {
  (void)in_sizes; (void)n_in; (void)out_size; (void)ws_size;
  const float* inp = (const float*)d_in[0];
  const float* U   = (const float*)d_in[1];
  const float* B0  = (const float*)d_in[7];
  const float* B1  = (const float*)d_in[13];
  float* out = (float*)d_out;

  float* ws = (float*)d_ws;
  size_t off = 0;
  auto alloc = [&](size_t n) -> float* { float* p = ws + off; off += n; return p; };

  float* AF   = alloc(4ull * kNP * kNP);
  float* ATMP = alloc((size_t)kN * kN * kR);
  const int wK[10]   = {kF, kF, kF, kH, kH, kH, kH, kH, kH, kH};
  const int widx[10] = {2, 3, 4, 5, 6, 8, 9, 10, 11, 12};
  float* WF[10];
  for (int i = 0; i < 10; ++i) WF[i] = alloc(4ull * wK[i] * kH);
  const size_t FB = 4ull * kNP * kWH;   // face buffer (max width)
  float* XF = alloc(FB); float* HF = alloc(FB);
  float* AX = alloc(FB); float* AH = alloc(FB);
  float* GF = alloc(FB); float* AG = alloc(FB);
  float* ZP = alloc(FB); float* RP = alloc(FB); float* HP = alloc(FB);
  const size_t SB = (size_t)kNP * kWH * 4;  // time-domain state buffer
  float* H0[2] = {alloc(SB), alloc(SB)};
  float* H1[2] = {alloc(SB), alloc(SB)};
  float* ZT = alloc(SB); float* GT = alloc(SB);

  Ctx c{stream, AF, HF, AX, AH, GF, AG, ZP, RP, HP, ZT, GT};

  // zero states every call (deterministic)
  k_zero<<<ewg(SB), 256, 0, stream>>>(H0[0], (int)SB);
  k_zero<<<ewg(SB), 256, 0, stream>>>(H0[1], (int)SB);
  k_zero<<<ewg(SB), 256, 0, stream>>>(H1[0], (int)SB);
  k_zero<<<ewg(SB), 256, 0, stream>>>(H1[1], (int)SB);

  // adaptive adjacency -> faces
  k_adj_tprod<<<ewg((size_t)kN * kN), 256, 0, stream>>>(U, ATMP);
  k_adj_softmax<<<ewg((size_t)kN * kR), 256, 0, stream>>>(ATMP);
  k_adj_fft<<<ewg((size_t)kNP * kNP), 256, 0, stream>>>(ATMP, AF);

  // weight faces
  for (int i = 0; i < 10; ++i)
    k_w_fft<<<ewg((size_t)wK[i] * kH), 256, 0, stream>>>(
        (const float*)d_in[widx[i]], WF[i], wK[i] * kH);

  const size_t EW = (size_t)kNP * kWH;

  // t = 0: layer1 consumes current layer0 output
  k_in_fft<<<ewg((size_t)kNP * kWX), 256, 0, stream>>>(inp, 0, XF);
  run_cell<kWX, kF>(c, XF, WF[0], WF[1], WF[2], WF[3], WF[4], B0, H0[0], H0[1],
                    nullptr, 0);
  k_s_fft<<<ewg(EW), 256, 0, stream>>>(H0[1], XF, kWH);
  run_cell<kWH, kH>(c, XF, WF[5], WF[6], WF[7], WF[8], WF[9], B1, H1[0], H1[1],
                    out, 0);

  int c0 = 1, c1 = 1;
  for (int t = 1; t < kT; ++t) {
    // layer1 first: input = previous layer0 output (p0), hidden = p1
    k_s_fft<<<ewg(EW), 256, 0, stream>>>(H0[c0], XF, kWH);
    run_cell<kWH, kH>(c, XF, WF[5], WF[6], WF[7], WF[8], WF[9], B1, H1[c1],
                      H1[1 - c1], out, t);
    // then layer0: input = inputs[:, t], hidden = p0
    k_in_fft<<<ewg((size_t)kNP * kWX), 256, 0, stream>>>(inp, t, XF);
    run_cell<kWX, kF>(c, XF, WF[0], WF[1], WF[2], WF[3], WF[4], B0, H0[c0],
                      H0[1 - c0], nullptr, t);
    c0 = 1 - c0;
    c1 = 1 - c1;
  }

  // h_last = stack([c0_state, c1_state]) appended after out
  const size_t outOff = (size_t)kB * kT * kN * kH * kR;
  const size_t lay = (size_t)kB * kN * kH * kR;
  k_copy_hlast<<<ewg((size_t)kB * kN * kH), 256, 0, stream>>>(H0[c0], out + outOff);
  k_copy_hlast<<<ewg((size_t)kB * kN * kH), 256, 0, stream>>>(H1[c1],
                                                              out + outOff + lay);
}